// DrugEmbed_35734127903521
// MI455X (gfx1250) — compile-verified
//
#include <hip/hip_runtime.h>
#include <hip/hip_bf16.h>
#include <math.h>

typedef __attribute__((ext_vector_type(2))) float v2f;
typedef __attribute__((ext_vector_type(8))) float v8f;

// ---------------------------------------------------------------------------
// Utility kernels
// ---------------------------------------------------------------------------
__global__ void k_fill(float* __restrict__ p, float v, size_t n) {
    size_t t = (size_t)blockIdx.x * blockDim.x + threadIdx.x;
    if (t < n) p[t] = v;
}

__global__ void k_deg_edges(const int* __restrict__ dst, float* __restrict__ deg, int E) {
    int t = blockIdx.x * blockDim.x + threadIdx.x;
    if (t < E) atomicAdd(&deg[dst[t]], 1.0f);
}

__global__ void k_rsqrt_inplace(float* __restrict__ p, int n) {
    int t = blockIdx.x * blockDim.x + threadIdx.x;
    if (t < n) p[t] = rsqrtf(p[t]);   // deg >= 1 always (self loops)
}

// ---------------------------------------------------------------------------
// WMMA GEMM:  H[nrows, NC] = A[nrows,128] @ W[128, NC]
// Block = 256 threads = 8 waves; each wave computes a 16-row strip.
// W staged in LDS (zero-padded to NT*16 cols). A fragments preloaded to VGPRs
// once per wave; B fragments preloaded to VGPRs once per column tile so the
// 32-step K accumulation runs as back-to-back v_wmma with a single LDS wait.
// Fragment layouts per CDNA5 ISA 7.12.2:
//   A 16x4 f32 : lane L -> M=L%16, ks=2*(L/16); v2f = {A[M][ks], A[M][ks+1]}
//   B 4x16 f32 : lane L -> N=L%16, ks=2*(L/16); v2f = {B[ks][N], B[ks+1][N]}
//   C/D 16x16  : lane L, reg r -> D[r + 8*(L/16)][L%16]
// ---------------------------------------------------------------------------
template<int NC, int NT>
__global__ __launch_bounds__(256) void k_gemm_wmma(const float* __restrict__ A,
                                                   const float* __restrict__ W,
                                                   float* __restrict__ H, int nrows) {
    constexpr int NPAD = NT * 16;
    __shared__ float sW[128 * NPAD];

    int tid = threadIdx.x;
    for (int i = tid; i < 128 * NPAD; i += 256) {
        int k = i / NPAD, c = i - k * NPAD;
        sW[i] = (c < NC) ? W[k * NC + c] : 0.0f;
    }
    __syncthreads();

    int wave = tid >> 5;
    int lane = tid & 31;
    int lm = lane & 15;       // M (A) / N (B,D) within tile
    int lh = lane >> 4;       // K-pair selector / row-half selector

    long rowBase = (long)blockIdx.x * 128 + wave * 16;
    long mrow = rowBase + lm;
    const float* arow = A + (size_t)(mrow < nrows ? mrow : 0) * 128;

    v2f afrag[32];
    #pragma unroll
    for (int k = 0; k < 32; ++k)
        afrag[k] = *(const v2f*)(arow + 4 * k + 2 * lh);

    for (int n = 0; n < NT; ++n) {
        const float* wcol = sW + n * 16 + lm;

        // Stage the whole tile's B fragments in VGPRs (64 regs), then run the
        // K-chain without per-step LDS dependencies.
        v2f bfrag[32];
        #pragma unroll
        for (int k = 0; k < 32; ++k) {
            int kk = 4 * k + 2 * lh;
            bfrag[k].x = wcol[kk * NPAD];
            bfrag[k].y = wcol[(kk + 1) * NPAD];
        }

        v8f acc = {0.f, 0.f, 0.f, 0.f, 0.f, 0.f, 0.f, 0.f};
        #pragma unroll
        for (int k = 0; k < 32; ++k)
            acc = __builtin_amdgcn_wmma_f32_16x16x4_f32(
                false, afrag[k], false, bfrag[k], (short)0, acc, false, false);

        int col = n * 16 + lm;
        if (col < NC) {
            #pragma unroll
            for (int r = 0; r < 8; ++r) {
                long orow = rowBase + r + 8 * lh;
                if (orow < nrows) H[(size_t)orow * NC + col] = acc[r];
            }
        }
    }
}

// ---------------------------------------------------------------------------
// out[i,:] = bias + h[i,:] * dinv[i]^2   (self-loop contribution + bias)
// ---------------------------------------------------------------------------
template<int F>
__global__ void k_selfloop_bias(const float* __restrict__ h, const float* __restrict__ dinv,
                                const float* __restrict__ bias, float* __restrict__ out, int n) {
    size_t t = (size_t)blockIdx.x * blockDim.x + threadIdx.x;
    size_t total = (size_t)n * F;
    if (t >= total) return;
    size_t i = t / F;
    int c = (int)(t - i * F);
    float di = dinv[i];
    out[t] = bias[c] + h[t] * di * di;
}

// ---------------------------------------------------------------------------
// Edge scatter: one warp per edge; lane handles 4 features (float4 gather,
// 4x global_atomic_add_f32 scatter).
// ---------------------------------------------------------------------------
template<int F, int LANES>
__global__ void k_scatter(const float* __restrict__ h, const float* __restrict__ dinv,
                          const int* __restrict__ src, const int* __restrict__ dst,
                          float* __restrict__ out, int E) {
    long t = (long)blockIdx.x * blockDim.x + threadIdx.x;
    int e = (int)(t >> 5);
    int lane = (int)(t & 31);
    if (e >= E || lane >= LANES) return;
    int s = src[e], d = dst[e];
    float norm = dinv[s] * dinv[d];
    float4 v = *(const float4*)(h + (size_t)s * F + lane * 4);
    float* o = out + (size_t)d * F + lane * 4;
    atomicAdd(o + 0, v.x * norm);
    atomicAdd(o + 1, v.y * norm);
    atomicAdd(o + 2, v.z * norm);
    atomicAdd(o + 3, v.w * norm);
}

// ---------------------------------------------------------------------------
// BatchNorm: per-column sum / sum-of-squares via LDS atomics, then global
// ---------------------------------------------------------------------------
__global__ void k_bn_partial(const float* __restrict__ x, float* __restrict__ stats, size_t total) {
    __shared__ float ssum[128];
    __shared__ float ssq[128];
    int tid = threadIdx.x;
    if (tid < 128) { ssum[tid] = 0.f; ssq[tid] = 0.f; }
    __syncthreads();
    size_t stride = (size_t)gridDim.x * blockDim.x;
    for (size_t i = (size_t)blockIdx.x * blockDim.x + tid; i < total; i += stride) {
        float v = x[i];
        int c = (int)(i & 127);
        atomicAdd(&ssum[c], v);
        atomicAdd(&ssq[c], v * v);
    }
    __syncthreads();
    if (tid < 128) {
        atomicAdd(&stats[tid], ssum[tid]);
        atomicAdd(&stats[128 + tid], ssq[tid]);
    }
}

__global__ void k_bn_finalize(float* __restrict__ stats, const float* __restrict__ gamma,
                              const float* __restrict__ beta, float invN) {
    int c = threadIdx.x;   // 128 threads
    float mu = stats[c] * invN;
    float var = stats[128 + c] * invN - mu * mu;
    float sc = gamma[c] * rsqrtf(var + 1e-5f);
    stats[256 + c] = sc;
    stats[384 + c] = beta[c] - mu * sc;
}

__global__ void k_bn_apply(float* __restrict__ x, const float* __restrict__ stats, size_t total) {
    size_t t = (size_t)blockIdx.x * blockDim.x + threadIdx.x;
    if (t >= total) return;
    int c = (int)(t & 127);
    x[t] = fmaxf(fmaf(x[t], stats[256 + c], stats[384 + c]), 0.0f);
}

// ---------------------------------------------------------------------------
// Segment-max pool: atomic float max via signed/unsigned int trick.
// ---------------------------------------------------------------------------
__device__ __forceinline__ void atomicMaxF(float* addr, float v) {
    if (v >= 0.0f) atomicMax((int*)addr, __float_as_int(v));
    else           atomicMin((unsigned int*)addr, __float_as_uint(v));
}

__global__ void k_pool_max(const float* __restrict__ x, const int* __restrict__ batch,
                           float* __restrict__ out, int n) {
    long t = (long)blockIdx.x * blockDim.x + threadIdx.x;
    int i = (int)(t >> 5);
    int lane = (int)(t & 31);
    if (i >= n || lane >= 25) return;   // 25 lanes * 4 = 100 cols
    int g = batch[i];
    float4 v = *(const float4*)(x + (size_t)i * 100 + lane * 4);
    float* o = out + (size_t)g * 100 + lane * 4;
    atomicMaxF(o + 0, v.x);
    atomicMaxF(o + 1, v.y);
    atomicMaxF(o + 2, v.z);
    atomicMaxF(o + 3, v.w);
}

// ---------------------------------------------------------------------------
extern "C" void kernel_launch(void* const* d_in, const int* in_sizes, int n_in,
                              void* d_out, int out_size, void* d_ws, size_t ws_size,
                              hipStream_t stream) {
    const float* X     = (const float*)d_in[0];
    const int*   ei    = (const int*)d_in[1];
    const int*   batch = (const int*)d_in[2];
    const float* W1    = (const float*)d_in[4];
    const float* b1    = (const float*)d_in[5];
    const float* gamma = (const float*)d_in[6];
    const float* beta  = (const float*)d_in[7];
    const float* W2    = (const float*)d_in[8];
    const float* b2    = (const float*)d_in[9];

    const int N = in_sizes[0] / 128;
    const int E = in_sizes[1] / 2;
    const int* src = ei;
    const int* dst = ei + E;
    float* out = (float*)d_out;

    // Workspace layout
    auto alignup = [](size_t v) { return (v + 255) & ~(size_t)255; };
    char* w = (char*)d_ws;
    size_t off = 0;
    float* d_dinv  = (float*)(w + off); off = alignup(off + (size_t)N * 4);
    float* d_stats = (float*)(w + off); off = alignup(off + 512 * 4);
    float* d_h     = (float*)(w + off); off = alignup(off + (size_t)N * 128 * 4); // h1, later h2
    float* d_x     = (float*)(w + off);                                           // out1/xrelu, later out2

    const size_t tot1 = (size_t)N * 128;
    const size_t tot2 = (size_t)N * 100;
    const long scatterT = (long)E * 32;
    const long poolT = (long)N * 32;

    // 1) degrees -> dinv
    k_fill<<<(N + 255) / 256, 256, 0, stream>>>(d_dinv, 1.0f, (size_t)N);
    k_deg_edges<<<(E + 255) / 256, 256, 0, stream>>>(dst, d_dinv, E);
    k_rsqrt_inplace<<<(N + 255) / 256, 256, 0, stream>>>(d_dinv, N);

    // 2) h1 = X @ W1  (WMMA f32)
    k_gemm_wmma<128, 8><<<(N + 127) / 128, 256, 0, stream>>>(X, W1, d_h, N);

    // 3) conv1 aggregate: self-loop + bias init, then edge scatter
    k_selfloop_bias<128><<<(unsigned)((tot1 + 255) / 256), 256, 0, stream>>>(d_h, d_dinv, b1, d_x, N);
    k_scatter<128, 32><<<(unsigned)((scatterT + 255) / 256), 256, 0, stream>>>(d_h, d_dinv, src, dst, d_x, E);

    // 4) BatchNorm + ReLU (in place on d_x)
    k_fill<<<2, 256, 0, stream>>>(d_stats, 0.0f, 512);
    k_bn_partial<<<2048, 256, 0, stream>>>(d_x, d_stats, tot1);
    k_bn_finalize<<<1, 128, 0, stream>>>(d_stats, gamma, beta, 1.0f / (float)N);
    k_bn_apply<<<(unsigned)((tot1 + 255) / 256), 256, 0, stream>>>(d_x, d_stats, tot1);

    // 5) h2 = xrelu @ W2  (WMMA f32, 100 cols padded to 112 in LDS)
    k_gemm_wmma<100, 7><<<(N + 127) / 128, 256, 0, stream>>>(d_x, W2, d_h, N);

    // 6) conv2 aggregate (d_x reused as out2)
    k_selfloop_bias<100><<<(unsigned)((tot2 + 255) / 256), 256, 0, stream>>>(d_h, d_dinv, b2, d_x, N);
    k_scatter<100, 25><<<(unsigned)((scatterT + 255) / 256), 256, 0, stream>>>(d_h, d_dinv, src, dst, d_x, E);

    // 7) segment max pool
    k_fill<<<(out_size + 255) / 256, 256, 0, stream>>>(out, -INFINITY, (size_t)out_size);
    k_pool_max<<<(unsigned)((poolT + 255) / 256), 256, 0, stream>>>(d_x, batch, out, N);
}